// STPNCell_33157147525556
// MI455X (gfx1250) — compile-verified
//
#include <hip/hip_runtime.h>

#define B_   32
#define I_   512
#define H_   1024
#define X_   1536
#define XCH  4
#define XC_  (X_ / XCH)   // 384

typedef float v2f __attribute__((ext_vector_type(2)));
typedef float v4f __attribute__((ext_vector_type(4)));
typedef float v8f __attribute__((ext_vector_type(8)));

// ---------------------------------------------------------------------------
// xh = concat(input, h0) along the feature axis: [B, X]
// ---------------------------------------------------------------------------
__global__ void k_concat(const float* __restrict__ in, const float* __restrict__ h0,
                         float* __restrict__ xh) {
    int t = blockIdx.x * 256 + threadIdx.x;          // B*X = 49152 threads
    int b = t / X_, x = t - b * X_;
    xh[t] = (x < I_) ? in[b * I_ + x] : h0[b * H_ + (x - I_)];
}

// ---------------------------------------------------------------------------
// Streaming pass 1 over F0 (NT loads):
//   partN[c,b,h] = sum_{x in chunk c} (W[x,h]+F0[b,x,h])^2
//   partS[c,b,h] = sum_{x in chunk c} xh[b,x]*F0[b,x,h]
// Fully coalesced along h; xh slice staged in LDS.
// ---------------------------------------------------------------------------
__global__ void k_reduce(const float* __restrict__ F0, const float* __restrict__ W,
                         const float* __restrict__ xh,
                         float* __restrict__ partN, float* __restrict__ partS) {
    int bid = blockIdx.x;
    int b  = bid & 31;          // fastest: concurrent blocks share the W slice in L2
    int c  = (bid >> 5) & 3;    // x-chunk
    int hb = bid >> 7;          // h-chunk (0..3)
    int h  = hb * 256 + threadIdx.x;
    int x0 = c * XC_;

    __shared__ float sxh[XC_];
    for (int i = threadIdx.x; i < XC_; i += 256) sxh[i] = xh[b * X_ + x0 + i];
    __syncthreads();

    const float* f0p = F0 + ((size_t)b * X_ + x0) * H_ + h;
    const float* wp  = W + (size_t)x0 * H_ + h;
    float n2 = 0.f, s = 0.f;
#pragma unroll 4
    for (int i = 0; i < XC_; ++i) {
        float f = __builtin_nontemporal_load(f0p);   // streaming, don't pollute L2
        float w = *wp;                               // reused across b -> L2 resident
        float g = w + f;
        n2 = fmaf(g, g, n2);
        s  = fmaf(sxh[i], f, s);
        f0p += H_; wp += H_;
    }
    int o = (c * B_ + b) * H_ + h;
    partN[o] = n2;
    partS[o] = s;
}

// ---------------------------------------------------------------------------
// sW = xh[32x1536] @ W[1536x1024] via V_WMMA_F32_16X16X4_F32, then
// h1 = tanh((sW + sF0)/sqrt(norm2) + bias);   also emit rnorm = 1/sqrt(norm2).
// 128 waves, one 16x16 output tile each, K stepped by 4.
// ---------------------------------------------------------------------------
__global__ void k_gemm_h1(const float* __restrict__ xh, const float* __restrict__ W,
                          const float* __restrict__ bias,
                          const float* __restrict__ partN, const float* __restrict__ partS,
                          float* __restrict__ h1, float* __restrict__ rnorm) {
    int wave   = blockIdx.x * 4 + (threadIdx.x >> 5);
    int lane   = threadIdx.x & 31;
    int lane16 = lane & 15;
    int khalf  = lane >> 4;           // 0: K=0,1   1: K=2,3  (A/B VGPR layout)
    int mtile  = wave & 1;            // M = 32 -> 2 tiles
    int ntile  = wave >> 1;           // N = 1024 -> 64 tiles
    int m = mtile * 16 + lane16;
    int n = ntile * 16 + lane16;

    const v2f* arow = (const v2f*)(xh + (size_t)m * X_);
    v8f acc = {};
#pragma unroll 2
    for (int k = 0; k < X_; k += 4) {
        v2f a = arow[(k >> 1) + khalf];               // A[m, k+2*khalf .. +1]
        int kk = k + khalf * 2;
        v2f bm;
        bm.x = W[(size_t)kk * H_ + n];                // B[kk,   n]
        bm.y = W[(size_t)(kk + 1) * H_ + n];          // B[kk+1, n]
        acc = __builtin_amdgcn_wmma_f32_16x16x4_f32(
            /*neg_a=*/false, a, /*neg_b=*/false, bm,
            /*c_mod=*/(short)0, acc, /*reuse_a=*/false, /*reuse_b=*/false);
    }

#pragma unroll
    for (int r = 0; r < 8; ++r) {
        int bb = mtile * 16 + khalf * 8 + r;          // C/D layout: lanes>=16 hold M=8..15
        int hh = ntile * 16 + lane16;
        int o  = bb * H_ + hh;
        float n2 = partN[o] + partN[B_ * H_ + o] + partN[2 * B_ * H_ + o] + partN[3 * B_ * H_ + o];
        float sf = partS[o] + partS[B_ * H_ + o] + partS[2 * B_ * H_ + o] + partS[3 * B_ * H_ + o];
        float rn = rsqrtf(n2);
        float hv = tanhf((acc[r] + sf) * rn + bias[hh]);
        h1[o]    = hv;
        rnorm[o] = rn;
    }
}

// ---------------------------------------------------------------------------
// Streaming pass 2 (NT load F0, NT store F1):
//   F1 = Gamma * (xh ⊗ h1) + Lambda * F0 / norm
// float4 per thread; Gamma/Lambda/h1/rnorm stay hot in L2.
// ---------------------------------------------------------------------------
__global__ void k_f1(const float* __restrict__ F0, const float* __restrict__ Gamma,
                     const float* __restrict__ Lambda, const float* __restrict__ xh,
                     const float* __restrict__ h1, const float* __restrict__ rnorm,
                     float* __restrict__ F1) {
    size_t t = (size_t)blockIdx.x * 256 + threadIdx.x;   // over B*X*H/4 float4s
    int    h4   = (int)(t & 255);                        // H/4 = 256
    size_t rest = t >> 8;
    int    x = (int)(rest % X_);
    int    b = (int)(rest / X_);
    size_t e = t * 4;

    v4f f0 = __builtin_nontemporal_load((const v4f*)(F0 + e));
    v4f gm = *(const v4f*)(Gamma  + (size_t)x * H_ + h4 * 4);
    v4f lm = *(const v4f*)(Lambda + (size_t)x * H_ + h4 * 4);
    v4f hv = *(const v4f*)(h1     + (size_t)b * H_ + h4 * 4);
    v4f rn = *(const v4f*)(rnorm  + (size_t)b * H_ + h4 * 4);
    float xs = xh[b * X_ + x];

    v4f outv = gm * (xs * hv) + lm * f0 * rn;
    __builtin_nontemporal_store(outv, (v4f*)(F1 + e));
}

// ---------------------------------------------------------------------------
extern "C" void kernel_launch(void* const* d_in, const int* in_sizes, int n_in,
                              void* d_out, int out_size, void* d_ws, size_t ws_size,
                              hipStream_t stream) {
    (void)in_sizes; (void)n_in; (void)out_size; (void)ws_size;
    const float* input  = (const float*)d_in[0];
    const float* h0     = (const float*)d_in[1];
    const float* F0     = (const float*)d_in[2];
    const float* W      = (const float*)d_in[3];
    const float* bias   = (const float*)d_in[4];
    const float* Gamma  = (const float*)d_in[5];
    const float* Lambda = (const float*)d_in[6];

    float* out = (float*)d_out;
    float* h1  = out;                 // [B,H]   first output
    float* F1  = out + B_ * H_;       // [B,X,H] second output

    float* ws    = (float*)d_ws;
    float* xh    = ws;                          // B*X       = 49152
    float* partN = xh + B_ * X_;                // XCH*B*H   = 131072
    float* partS = partN + XCH * B_ * H_;       // XCH*B*H   = 131072
    float* rnorm = partS + XCH * B_ * H_;       // B*H       = 32768

    k_concat <<<(B_ * X_) / 256, 256, 0, stream>>>(input, h0, xh);
    k_reduce <<<B_ * XCH * (H_ / 256), 256, 0, stream>>>(F0, W, xh, partN, partS);
    k_gemm_h1<<<32, 128, 0, stream>>>(xh, W, bias, partN, partS, h1, rnorm);
    k_f1     <<<(int)(((size_t)B_ * X_ * H_ / 4) / 256), 256, 0, stream>>>(
                  F0, Gamma, Lambda, xh, h1, rnorm, F1);
}